// CausalAttentionLayer_89335319757377
// MI455X (gfx1250) — compile-verified
//
#include <hip/hip_runtime.h>
#include <math.h>

// ---- problem constants ----
#define B_   32
#define K_   64
#define W_   100
#define H_   150
#define H3   450
#define KW   6400
#define S_   64
#define KPIH 100     // padded K for wih (exact, even)
#define KPHH 152     // padded K for whh (150 -> 152, even)
#define NPAD 464     // padded gate dim (450 -> 29 tiles * 16)
#define GTS  36      // gate buffer [n][row] stride (16B-aligned, low conflict)
#define HTS  33      // LDS h pair-row stride (conflict-free)
#define LWPAD 112    // padded N for lin_w (100 -> 7 tiles * 16)

__device__ __constant__ float kALPHA = 0.2f;
__device__ __constant__ float kTHRES = 0.0002f;

typedef float v2f __attribute__((ext_vector_type(2)));
typedef float v4f __attribute__((ext_vector_type(4)));
typedef float v8f __attribute__((ext_vector_type(8)));

// D = A(16x4,f32) * B(4x16,f32) + C  -- exact fp32 matrix op on CDNA5
__device__ __forceinline__ v8f wmma_f32_4(v2f a, v2f b, v8f c) {
  return __builtin_amdgcn_wmma_f32_16x16x4_f32(
      false, a, false, b, (short)0, c, false, false);
}
__device__ __forceinline__ v2f ld2(const float* p) { return *(const v2f*)p; }

// ---------------------------------------------------------------
// Transpose + pad + pair-interleave weights:
//   logical dst[i][n] stored at dst[((i>>1)*npad + n)*2 + (i&1)]
// ---------------------------------------------------------------
__global__ void __launch_bounds__(256)
k_transpose_pad(const float* __restrict__ src, float* __restrict__ dst,
                int nsrc, int kd, int kdpad, int npad, int nmats, int src_stride) {
  const int per = kdpad * npad;
  const long total = (long)nmats * per;
  for (long idx = (long)blockIdx.x * blockDim.x + threadIdx.x; idx < total;
       idx += (long)gridDim.x * blockDim.x) {
    const int m   = (int)(idx / per);
    const int rem = (int)(idx % per);
    const int q  = rem & 1;
    const int r2 = rem >> 1;
    const int n  = r2 % npad;
    const int i  = (r2 / npad) * 2 + q;
    float v = 0.f;
    if (n < nsrc && i < kd) v = src[(long)m * src_stride + n * kd + i];
    dst[idx] = v;
  }
}

// x (2048 x 100) -> x2[((i>>1)*2048 + r)*2 + (i&1)]
__global__ void __launch_bounds__(256)
k_transX(const float* __restrict__ x, float* __restrict__ x2) {
  const int idx = blockIdx.x * blockDim.x + threadIdx.x;  // 204800
  if (idx >= 2048 * W_) return;
  const int q  = idx & 1;
  const int r2 = idx >> 1;
  const int r  = r2 % 2048;
  const int i  = (r2 / 2048) * 2 + q;
  x2[idx] = x[r * W_ + i];
}

// A[(b*64+s)*100 + i] -> AT[s*3200 + ((i>>1)*32 + b)*2 + (i&1)]
__global__ void __launch_bounds__(256)
k_transA(const float* __restrict__ src, float* __restrict__ dst) {
  const int idx = blockIdx.x * blockDim.x + threadIdx.x;  // 204800
  if (idx >= S_ * W_ * B_) return;
  const int s   = idx / (W_ * B_);
  const int rem = idx % (W_ * B_);
  const int q  = rem & 1;
  const int r2 = rem >> 1;
  const int b  = r2 & 31;
  const int i  = (r2 >> 5) * 2 + q;
  dst[idx] = src[(b * S_ + s) * W_ + i];
}

// ---------------------------------------------------------------
// Wx[r, w] = sum_i x[r,i] * lin_w[w,i] + lin_b[w]
// ---------------------------------------------------------------
__global__ void __launch_bounds__(256)
k_wx(const float* __restrict__ x2, const float* __restrict__ lin_w2,
     const float* __restrict__ lin_b, float* __restrict__ wx) {
  const int wave = blockIdx.x * (blockDim.x >> 5) + (threadIdx.x >> 5);
  const int NT_N = 7, NT_M = 128;
  if (wave >= NT_M * NT_N) return;
  const int tm = wave / NT_N, tn = wave % NT_N;
  const int lane = threadIdx.x & 31;
  const int half = lane >> 4;
  const int l15  = lane & 15;
  const int arow = tm * 16 + l15;     // < 2048
  const int ncol = tn * 16 + l15;     // < 112 (padded)
  const float* xp = x2 + ((size_t)half * 2048 + arow) * 2;
  const float* wp = lin_w2 + ((size_t)half * LWPAD + ncol) * 2;
  v8f acc = {};
  for (int kk = 0; kk < W_; kk += 4) {
    acc = wmma_f32_4(ld2(xp), ld2(wp), acc);
    xp += 2 * 2048 * 2;      // p0 += 2
    wp += 4 * LWPAD;
  }
  if (ncol < W_) {
    const float bias = lin_b[ncol];
#pragma unroll
    for (int r = 0; r < 8; ++r) {
      const int row = tm * 16 + r + half * 8;
      wx[row * W_ + ncol] = acc[r] + bias;
    }
  }
}

// ---------------------------------------------------------------
// Attention row per (b,k): leaky-relu softmax threshold sum
// ---------------------------------------------------------------
__global__ void __launch_bounds__(256)
k_attn(const float* __restrict__ y, const float* __restrict__ av,
       const float* __restrict__ bias, const float* __restrict__ wx,
       float* __restrict__ cause) {
  __shared__ float e_row[KW];
  __shared__ float red[256];
  const int bk = blockIdx.x;
  const int b = bk >> 6, k = bk & 63;
  const int t = threadIdx.x;
  const float a0 = av[0], a1 = av[1];
  const float yv = y[bk];
  const float* vrow = wx + b * KW;
  const float* brow = bias + k * KW;

  float mx = -3.4e38f;
  for (int j = t; j < KW; j += 256) {
    float e = a0 * yv + a1 * vrow[j] + brow[j];
    e = (e > 0.f) ? e : kALPHA * e;
    e_row[j] = e;
    mx = fmaxf(mx, e);
  }
  red[t] = mx; __syncthreads();
  for (int o = 128; o > 0; o >>= 1) { if (t < o) red[t] = fmaxf(red[t], red[t + o]); __syncthreads(); }
  mx = red[0]; __syncthreads();

  float sum = 0.f;
  for (int j = t; j < KW; j += 256) {
    const float ex = expf(e_row[j] - mx);
    e_row[j] = ex;
    sum += ex;
  }
  red[t] = sum; __syncthreads();
  for (int o = 128; o > 0; o >>= 1) { if (t < o) red[t] += red[t + o]; __syncthreads(); }
  sum = red[0]; __syncthreads();

  const float thr = kTHRES * sum;
  float s = 0.f;
  for (int j = t; j < KW; j += 256) if (e_row[j] >= thr) s += vrow[j];
  red[t] = s; __syncthreads();
  for (int o = 128; o > 0; o >>= 1) { if (t < o) red[t] += red[t + o]; __syncthreads(); }
  s = red[0];

  for (int w = t; w < W_; w += 256) cause[bk * W_ + w] = wx[bk * W_ + w] + s;
}

// ---------------------------------------------------------------
// GRU step GEMMs: one N-tile per wave, both M-tiles (B frags reused)
//   pointer-increment streams, b64 operand loads, b128 gate stores
//   gate buffers are [n][row] (GTS stride)
// ---------------------------------------------------------------
__device__ __forceinline__ void gru_step_gemm(
    const float* __restrict__ AxT,   // [s][pair-interleaved 100x32]
    int s,
    const float* __restrict__ wih2,  // pair-interleaved [100][464], zero-padded
    const float* __restrict__ whh2,  // pair-interleaved [152][464], zero-padded
    const float* __restrict__ bih, const float* __restrict__ bhh,
    const float* h2, float* gx_t, float* gh_t) {
  const int wid  = threadIdx.x >> 5;
  const int nw   = blockDim.x >> 5;
  const int lane = threadIdx.x & 31;
  const int half = lane >> 4;
  const int l15  = lane & 15;
  const float* As = AxT + s * (W_ * B_);
  for (int tn = wid; tn < NPAD / 16; tn += nw) {
    const int n = tn * 16 + l15;
    v8f g0 = {}, g1 = {}, hh0 = {}, hh1 = {};
    // gx: K = 100 exact
    {
      const float* wp = wih2 + ((size_t)half * NPAD + n) * 2;
      const float* ap = As + ((size_t)half * 32 + l15) * 2;
      for (int kk = 0; kk < KPIH; kk += 4) {
        const v2f b  = ld2(wp);
        const v2f a0 = ld2(ap);
        const v2f a1 = ld2(ap + 32);
        g0 = wmma_f32_4(a0, b, g0);
        g1 = wmma_f32_4(a1, b, g1);
        wp += 4 * NPAD;   // p0 += 2
        ap += 128;
      }
    }
    // gh: K = 152 (rows 150/151 are zeros in both h2 and whh2)
    {
      const float* wp = whh2 + ((size_t)half * NPAD + n) * 2;
      const float* hp = h2 + ((size_t)half * HTS + l15) * 2;
      for (int kk = 0; kk < KPHH; kk += 4) {
        const v2f b  = ld2(wp);
        const v2f a0 = ld2(hp);
        const v2f a1 = ld2(hp + 32);
        hh0 = wmma_f32_4(a0, b, hh0);
        hh1 = wmma_f32_4(a1, b, hh1);
        wp += 4 * NPAD;
        hp += 4 * HTS;
      }
    }
    const int nc = (n < H3) ? n : (H3 - 1);   // clamp, no branch
    const float bi = bih[nc], bh = bhh[nc];
    float* gx0 = gx_t + n * GTS + half * 8;   // rows half*8..+7 (m-tile 0)
    float* gh0 = gh_t + n * GTS + half * 8;
    *(v4f*)(gx0)      = (v4f){g0[0] + bi, g0[1] + bi, g0[2] + bi, g0[3] + bi};
    *(v4f*)(gx0 + 4)  = (v4f){g0[4] + bi, g0[5] + bi, g0[6] + bi, g0[7] + bi};
    *(v4f*)(gx0 + 16) = (v4f){g1[0] + bi, g1[1] + bi, g1[2] + bi, g1[3] + bi};
    *(v4f*)(gx0 + 20) = (v4f){g1[4] + bi, g1[5] + bi, g1[6] + bi, g1[7] + bi};
    *(v4f*)(gh0)      = (v4f){hh0[0] + bh, hh0[1] + bh, hh0[2] + bh, hh0[3] + bh};
    *(v4f*)(gh0 + 4)  = (v4f){hh0[4] + bh, hh0[5] + bh, hh0[6] + bh, hh0[7] + bh};
    *(v4f*)(gh0 + 16) = (v4f){hh1[0] + bh, hh1[1] + bh, hh1[2] + bh, hh1[3] + bh};
    *(v4f*)(gh0 + 20) = (v4f){hh1[4] + bh, hh1[5] + bh, hh1[6] + bh, hh1[7] + bh};
  }
}

__device__ __forceinline__ int hidx(int j, int bb) {  // h pair-interleave
  return ((j >> 1) * HTS + bb) * 2 + (j & 1);
}

__device__ __forceinline__ void gru_step_elem(
    float* h2, const float* gx_t, const float* gh_t) {
  for (int idx = threadIdx.x; idx < B_ * H_; idx += blockDim.x) {
    const int bb = idx / H_, j = idx % H_;   // lanes sweep j
    const float r  = 1.f / (1.f + expf(-(gx_t[j * GTS + bb] + gh_t[j * GTS + bb])));
    const float zg = 1.f / (1.f + expf(-(gx_t[(H_ + j) * GTS + bb] + gh_t[(H_ + j) * GTS + bb])));
    const float nn = tanhf(gx_t[(2 * H_ + j) * GTS + bb] + r * gh_t[(2 * H_ + j) * GTS + bb]);
    const int hi = hidx(j, bb);
    h2[hi] = (1.f - zg) * nn + zg * h2[hi];
  }
}

// ---------------------------------------------------------------
// Global GRU over cause_x, then mu / log_var / z heads
// ---------------------------------------------------------------
__global__ void __launch_bounds__(256)
k_gru_global(const float* __restrict__ causeT,
             const float* __restrict__ wih2, const float* __restrict__ whh2,
             const float* __restrict__ bih, const float* __restrict__ bhh,
             const float* __restrict__ mu_w, const float* __restrict__ mu_b,
             const float* __restrict__ std_w, const float* __restrict__ std_b,
             const float* __restrict__ z_noise, float* __restrict__ z_out) {
  __shared__ __align__(16) float h2[(KPHH / 2) * HTS * 2];
  __shared__ __align__(16) float gx_t[NPAD * GTS];
  __shared__ __align__(16) float gh_t[NPAD * GTS];
  for (int i = threadIdx.x; i < (KPHH / 2) * HTS * 2; i += blockDim.x) h2[i] = 0.f;
  __syncthreads();
  for (int s = 0; s < S_; ++s) {
    gru_step_gemm(causeT, s, wih2, whh2, bih, bhh, h2, gx_t, gh_t);
    __syncthreads();
    gru_step_elem(h2, gx_t, gh_t);
    __syncthreads();
  }
  for (int idx = threadIdx.x; idx < B_ * H_; idx += blockDim.x) {
    const int bb = idx / H_, j = idx % H_;
    float mu = mu_b[j], lv = std_b[j];
    for (int tt = 0; tt < H_; ++tt) {
      const float hv = h2[hidx(tt, bb)];
      mu += hv * mu_w[j * H_ + tt];
      lv += hv * std_w[j * H_ + tt];
    }
    z_out[idx] = mu + expf(0.5f * lv) * z_noise[idx];
  }
}

// ---------------------------------------------------------------
// 64 independent per-k GRUs over Wx, h0 = z  ->  out (B,K,H)
// ---------------------------------------------------------------
__global__ void __launch_bounds__(256)
k_gru_nets(const float* __restrict__ wxT,
           const float* __restrict__ wih2_all, const float* __restrict__ whh2_all,
           const float* __restrict__ net_bih, const float* __restrict__ net_bhh,
           const float* __restrict__ z_in, float* __restrict__ out) {
  __shared__ __align__(16) float h2[(KPHH / 2) * HTS * 2];
  __shared__ __align__(16) float gx_t[NPAD * GTS];
  __shared__ __align__(16) float gh_t[NPAD * GTS];
  const int k = blockIdx.x;
  const float* wih2 = wih2_all + (size_t)k * KPIH * NPAD;
  const float* whh2 = whh2_all + (size_t)k * KPHH * NPAD;
  const float* bih = net_bih + k * H3;
  const float* bhh = net_bhh + k * H3;
  for (int i = threadIdx.x; i < (KPHH / 2) * HTS * 2; i += blockDim.x) h2[i] = 0.f;
  __syncthreads();
  for (int idx = threadIdx.x; idx < B_ * H_; idx += blockDim.x) {
    const int j = idx >> 5, bb = idx & 31;
    h2[hidx(j, bb)] = z_in[bb * H_ + j];
  }
  __syncthreads();
  for (int s = 0; s < S_; ++s) {
    gru_step_gemm(wxT, s, wih2, whh2, bih, bhh, h2, gx_t, gh_t);
    __syncthreads();
    gru_step_elem(h2, gx_t, gh_t);
    __syncthreads();
  }
  for (int idx = threadIdx.x; idx < B_ * H_; idx += blockDim.x) {
    const int bb = idx / H_, j = idx % H_;
    out[(bb * K_ + k) * H_ + j] = h2[hidx(j, bb)];
  }
}

// ---------------------------------------------------------------
extern "C" void kernel_launch(void* const* d_in, const int* in_sizes, int n_in,
                              void* d_out, int out_size, void* d_ws, size_t ws_size,
                              hipStream_t stream) {
  const float* x       = (const float*)d_in[0];
  const float* y       = (const float*)d_in[1];
  const float* z_noise = (const float*)d_in[2];
  const float* lin_w   = (const float*)d_in[3];
  const float* lin_b   = (const float*)d_in[4];
  const float* a       = (const float*)d_in[5];
  const float* bias    = (const float*)d_in[6];
  const float* gl_wih  = (const float*)d_in[7];
  const float* gl_whh  = (const float*)d_in[8];
  const float* gl_bih  = (const float*)d_in[9];
  const float* gl_bhh  = (const float*)d_in[10];
  const float* mu_w    = (const float*)d_in[11];
  const float* mu_b    = (const float*)d_in[12];
  const float* std_w   = (const float*)d_in[13];
  const float* std_b   = (const float*)d_in[14];
  const float* net_wih = (const float*)d_in[15];
  const float* net_whh = (const float*)d_in[16];
  const float* net_bih = (const float*)d_in[17];
  const float* net_bhh = (const float*)d_in[18];

  float* ws = (float*)d_ws;
  float* wx      = ws;                 // 204800
  float* cause   = ws + 204800;        // 204800
  float* z       = ws + 409600;        // 4800
  float* wxT     = ws + 414400;        // 204800
  float* causeT  = ws + 619200;        // 204800
  float* xT      = ws + 824000;        // 204800
  float* lin_wT  = ws + 1028800;       // 100*112   = 11200
  float* glwihT  = ws + 1040000;       // 100*464   = 46400
  float* glwhhT  = ws + 1086400;       // 152*464   = 70528
  float* nwihT   = ws + 1156928;       // 64*100*464 = 2969600
  float* nwhhT   = ws + 4126528;       // 64*152*464 = 4513792
  float* out = (float*)d_out;

  // weight transposes (coalesced, zero-padded, pair-interleaved B operands)
  k_transpose_pad<<<(100 * LWPAD + 255) / 256, 256, 0, stream>>>(
      lin_w, lin_wT, 100, 100, 100, LWPAD, 1, 100 * 100);
  k_transpose_pad<<<(KPIH * NPAD + 255) / 256, 256, 0, stream>>>(
      gl_wih, glwihT, H3, W_, KPIH, NPAD, 1, H3 * W_);
  k_transpose_pad<<<(KPHH * NPAD + 255) / 256, 256, 0, stream>>>(
      gl_whh, glwhhT, H3, H_, KPHH, NPAD, 1, H3 * H_);
  k_transpose_pad<<<4096, 256, 0, stream>>>(
      net_wih, nwihT, H3, W_, KPIH, NPAD, K_, H3 * W_);
  k_transpose_pad<<<4096, 256, 0, stream>>>(
      net_whh, nwhhT, H3, H_, KPHH, NPAD, K_, H3 * H_);
  k_transX<<<800, 256, 0, stream>>>(x, xT);

  k_wx<<<112, 256, 0, stream>>>(xT, lin_wT, lin_b, wx);
  k_attn<<<2048, 256, 0, stream>>>(y, a, bias, wx, cause);
  k_transA<<<800, 256, 0, stream>>>(wx, wxT);
  k_transA<<<800, 256, 0, stream>>>(cause, causeT);

  k_gru_global<<<1, 256, 0, stream>>>(causeT, glwihT, glwhhT, gl_bih, gl_bhh,
                                      mu_w, mu_b, std_w, std_b, z_noise, z);
  k_gru_nets<<<64, 256, 0, stream>>>(wxT, nwihT, nwhhT, net_bih, net_bhh, z, out);
}